// TriadNodeNet_59734405153437
// MI455X (gfx1250) — compile-verified
//
#include <hip/hip_runtime.h>
#include <math.h>

typedef float v2f __attribute__((ext_vector_type(2)));
typedef float v4f __attribute__((ext_vector_type(4)));
typedef float v8f __attribute__((ext_vector_type(8)));

#define B_    4
#define N_    1024
#define D_    128
#define INDIM 8192
#define NUMIN 64
#define EPS_  1e-8f
#define CMOM  0.99f
#define AVAR  0.02f
#define ITERS 8
#define RB_   16                          // column-normalize row chunks (64 rows each)
#define BUNT_SCALE 1.8477590650225735f    // sqrt(2+sqrt(2))
#define BASELINE   (1.0f/1025.0f)         // 1/(N+1)
#define INV_SQRT_D 0.08838834764831845f   // 1/sqrt(128)

// workspace layout (floats)
#define Q_OFF   ((size_t)0)
#define K_OFF   ((size_t)(B_*N_*D_))               // 524288
#define VT_OFF  ((size_t)(2*B_*N_*D_))             // 1048576  (V transposed: [B,D,N])
#define M_OFF   ((size_t)(3*B_*N_*D_))             // 1572864, size B*N*N
#define P_OFF   (M_OFF + (size_t)B_*N_*N_)         // col partials: B*RB_*N = 65536
#define SC_OFF  (P_OFF + (size_t)B_*RB_*N_)        // 16 scalars: ms[0..3], vs[4..7], vsq[8..11]

// d_out layout: [B,N,2D] then A [B,N,N]
#define OUT_A_OFF ((size_t)(B_*N_*2*D_))           // 1048576

__device__ __forceinline__ float buntanh(float x) {
    return BUNT_SCALE * (0.5f * x + 0.5f * tanhf(x));
}

// ---------------- init scalars ----------------
__global__ void k_init(float* scal) {
    if (threadIdx.x < 16) scal[threadIdx.x] = 0.0f;
}

// ---------------- per-node GEMVs + buntanh ----------------
// One wave per node; each lane owns 4 consecutive k columns -> all weight
// traffic is global_load_b128, fully coalesced 512B/row/matrix per wave.
__global__ void k_gemv(const float* __restrict__ state,
                       const float* __restrict__ w1,
                       const float* __restrict__ w2,
                       const float* __restrict__ w3,
                       float* __restrict__ Q, float* __restrict__ K,
                       float* __restrict__ out) {
    const int node = blockIdx.x;           // b*N + n
    const int t = threadIdx.x;             // 0..31
    __shared__ float s[D_];
    ((v4f*)s)[t] = ((const v4f*)(state + (size_t)node * D_))[t];
    __syncthreads();

    const size_t wbase = (size_t)node * D_ * D_ + (size_t)t * 4;
    const float* __restrict__ p1 = w1 + wbase;
    const float* __restrict__ p2 = w2 + wbase;
    const float* __restrict__ p3 = w3 + wbase;

    v4f a1 = {0.f,0.f,0.f,0.f}, a2 = {0.f,0.f,0.f,0.f}, a3 = {0.f,0.f,0.f,0.f};
#pragma unroll 2
    for (int d = 0; d < D_; ++d) {
        const float sd = s[d];
        const size_t o = (size_t)d * D_;
        const v4f x1 = *(const v4f*)(p1 + o);
        const v4f x2 = *(const v4f*)(p2 + o);
        const v4f x3 = *(const v4f*)(p3 + o);
        a1 += sd * x1;
        a2 += sd * x2;
        a3 += sd * x3;
    }

    v4f q, kk, pr;
#pragma unroll
    for (int j = 0; j < 4; ++j) {
        q[j]  = buntanh(a3[j]);    // w3 -> Q
        kk[j] = buntanh(a2[j]);    // w2 -> K
        pr[j] = buntanh(a1[j]);    // w1 -> prediction
    }
    *(v4f*)(Q + (size_t)node * D_ + t * 4) = q;
    *(v4f*)(K + (size_t)node * D_ + t * 4) = kk;
    *(v4f*)(out + (size_t)node * (2 * D_) + D_ + t * 4) = pr;
}

// ---------------- raw_A = Q K^T / sqrt(D) via v_wmma_f32_16x16x4_f32 ----------------
// One wave -> one 16x16 tile; fused per-batch sum-of-squares for rms_norm.
__global__ void k_qk_wmma(const float* __restrict__ Q, const float* __restrict__ K,
                          float* __restrict__ M, float* __restrict__ ms) {
    const int blk = blockIdx.x;
    const int b  = blk / (64 * 64);
    const int r  = blk % (64 * 64);
    const int ti = r / 64, tj = r % 64;
    const int lane = threadIdx.x;
    const int lr = lane & 15, lh = lane >> 4;

    const float* qrow = Q + ((size_t)b * N_ + ti * 16 + lr) * D_;
    const float* krow = K + ((size_t)b * N_ + tj * 16 + lr) * D_;

    v8f c = {0.f, 0.f, 0.f, 0.f, 0.f, 0.f, 0.f, 0.f};
#pragma unroll 4
    for (int k0 = 0; k0 < D_; k0 += 4) {
        v2f a  = *(const v2f*)(qrow + k0 + 2 * lh);   // A: 16x4 f32 fragment
        v2f bb = *(const v2f*)(krow + k0 + 2 * lh);   // B: (K^T) 4x16 fragment
        c = __builtin_amdgcn_wmma_f32_16x16x4_f32(false, a, false, bb,
                                                  (short)0, c, false, false);
    }

    float ssq = 0.f;
    float* Mb = M + (size_t)b * N_ * N_;
#pragma unroll
    for (int rr = 0; rr < 8; ++rr) {
        const float v = c[rr] * INV_SQRT_D;
        const int row = ti * 16 + rr + 8 * lh;
        const int col = tj * 16 + lr;
        Mb[(size_t)row * N_ + col] = v;
        ssq = fmaf(v, v, ssq);
    }
    for (int off = 16; off >= 1; off >>= 1) ssq += __shfl_xor(ssq, off, 32);
    if (lane == 0) atomicAdd(&ms[b], ssq);
}

// ---------------- rms_norm + EMA + mask + sinkhorn prep ----------------
__global__ void k_ema(const float* __restrict__ A_ema, const float* __restrict__ scal,
                      float* __restrict__ M) {
    const size_t total = (size_t)B_ * N_ * N_;
    for (size_t idx = (size_t)blockIdx.x * blockDim.x + threadIdx.x; idx < total;
         idx += (size_t)gridDim.x * blockDim.x) {
        const int b = (int)(idx / ((size_t)N_ * N_));
        const int r = (int)((idx / N_) % N_);
        const int c = (int)(idx % N_);
        float ema = A_ema[idx];
        if (r == NUMIN && c < NUMIN) ema += 3.0f / (float)NUMIN;   // first-call adjust
        const float msmean = scal[b] / (float)((size_t)N_ * N_);
        const float rn = M[idx] * rsqrtf(msmean + EPS_);
        float v = ema * CMOM + rn * (1.0f - CMOM);
        if (r < NUMIN) v = 0.f;                                    // A_mask
        M[idx] = fmaxf(v, 0.f) + EPS_;                             // sinkhorn entry
    }
}

// ---------------- sinkhorn: row normalize (block per row) ----------------
__global__ void k_rownorm(float* __restrict__ M) {
    float* p = M + (size_t)blockIdx.x * N_;
    const int t = threadIdx.x;
    __shared__ float red[256];
    float v0 = p[t], v1 = p[t + 256], v2 = p[t + 512], v3 = p[t + 768];
    red[t] = (v0 + v1) + (v2 + v3);
    __syncthreads();
    for (int s = 128; s > 0; s >>= 1) {
        if (t < s) red[t] += red[t + s];
        __syncthreads();
    }
    const float inv = 1.0f / red[0];
    p[t] = v0 * inv; p[t + 256] = v1 * inv; p[t + 512] = v2 * inv; p[t + 768] = v3 * inv;
}

// ---------------- sinkhorn col normalize, phase A: partial column sums ----------------
// grid = B*RB_*4 blocks of 256 threads; block sums a 64-row x 256-col chunk.
__global__ void k_colpart(const float* __restrict__ M, float* __restrict__ part) {
    const int blk = blockIdx.x;
    const int b  = blk / (RB_ * 4);
    const int rb = (blk / 4) % RB_;
    const int col = (blk % 4) * 256 + threadIdx.x;
    const float* p = M + (size_t)b * N_ * N_ + (size_t)rb * (N_ / RB_) * N_ + col;
    float sum = 0.f;
#pragma unroll 4
    for (int r = 0; r < N_ / RB_; ++r) sum += p[(size_t)r * N_];
    part[((size_t)b * RB_ + rb) * N_ + col] = sum;
}

// ---------------- sinkhorn col normalize, phase B: reduce partials + scale ----------------
__global__ void k_colscale(float* __restrict__ M, const float* __restrict__ part) {
    const int blk = blockIdx.x;
    const int b  = blk / (RB_ * 4);
    const int rb = (blk / 4) % RB_;
    const int col = (blk % 4) * 256 + threadIdx.x;
    float tot = 0.f;
#pragma unroll
    for (int j = 0; j < RB_; ++j) tot += part[((size_t)b * RB_ + j) * N_ + col];
    const float inv = 1.0f / tot;
    float* p = M + (size_t)b * N_ * N_ + (size_t)rb * (N_ / RB_) * N_ + col;
#pragma unroll 4
    for (int r = 0; r < N_ / RB_; ++r) p[(size_t)r * N_] *= inv;
}

// ---------------- threshold + second sinkhorn prep ----------------
__global__ void k_thresh(float* __restrict__ M) {
    const size_t total = (size_t)B_ * N_ * N_;
    for (size_t idx = (size_t)blockIdx.x * blockDim.x + threadIdx.x; idx < total;
         idx += (size_t)gridDim.x * blockDim.x) {
        const float v = M[idx];
        M[idx] = ((v > BASELINE) ? v : 0.f) + EPS_;   // sparse_A, then max(.,0)+EPS
    }
}

// ---------------- final A = M*mask -> d_out, fused variance sums ----------------
__global__ void k_finalA(const float* __restrict__ M, float* __restrict__ Aout,
                         float* __restrict__ scal) {
    const int rowIdx = blockIdx.x;
    const int b = rowIdx / N_, r = rowIdx % N_;
    const float mask = (r < NUMIN) ? 0.f : 1.f;
    const float* src = M + (size_t)rowIdx * N_;
    float* dst = Aout + (size_t)rowIdx * N_;
    const int t = threadIdx.x;
    __shared__ float rs[256], rq[256];
    float s = 0.f, q = 0.f;
#pragma unroll
    for (int j = 0; j < 4; ++j) {
        const float v = src[t + j * 256] * mask;
        dst[t + j * 256] = v;
        s += v; q = fmaf(v, v, q);
    }
    rs[t] = s; rq[t] = q;
    __syncthreads();
    for (int st = 128; st > 0; st >>= 1) {
        if (t < st) { rs[t] += rs[t + st]; rq[t] += rq[t + st]; }
        __syncthreads();
    }
    if (t == 0) { atomicAdd(&scal[4 + b], rs[0]); atomicAdd(&scal[8 + b], rq[0]); }
}

// ---------------- VT[b][d][n] = output + relu(a_var - var)*noise (LDS transpose) ----------------
// grid = B*32*4 blocks (32x32 tiles), block = (32,8)
__global__ void k_vt(const float* __restrict__ output, const float* __restrict__ noise,
                     const float* __restrict__ scal, float* __restrict__ VT) {
    const int blk = blockIdx.x;
    const int b  = blk / (32 * 4);
    const int tn = (blk % (32 * 4)) / 4;
    const int td = blk % 4;
    const int n0 = tn * 32, d0 = td * 32;
    const int tx = threadIdx.x, ty = threadIdx.y;

    const float cnt = (float)((size_t)N_ * N_);
    const float sum = scal[4 + b], sq = scal[8 + b];
    const float var = (sq - sum * sum / cnt) / (cnt - 1.0f);   // unbiased (ddof=1)
    const float vd = fmaxf(AVAR - var, 0.f);

    __shared__ float tile[32][33];
#pragma unroll
    for (int j = 0; j < 4; ++j) {
        const int n = n0 + ty + j * 8;
        const size_t idx = ((size_t)b * N_ + n) * D_ + d0 + tx;
        tile[ty + j * 8][tx] = fmaf(vd, noise[idx], output[idx]);   // coalesced read
    }
    __syncthreads();
#pragma unroll
    for (int j = 0; j < 4; ++j) {
        const int d = d0 + ty + j * 8;
        VT[((size_t)b * D_ + d) * N_ + n0 + tx] = tile[tx][ty + j * 8]; // coalesced write
    }
}

// ---------------- target = softsign(A @ V) via v_wmma_f32_16x16x4_f32 ----------------
// One wave -> one 16x16 tile, K=1024; A and VT fragments are both b64 loads.
__global__ void k_av_wmma(const float* __restrict__ Aout, const float* __restrict__ VT,
                          float* __restrict__ out) {
    const int blk = blockIdx.x;
    const int b = blk / (64 * 8);
    const int r = blk % (64 * 8);
    const int tn = r / 8, td = r % 8;
    const int lane = threadIdx.x;
    const int lr = lane & 15, lh = lane >> 4;

    const float* arow = Aout + ((size_t)b * N_ + tn * 16 + lr) * N_;
    const int dcol = td * 16 + lr;
    const float* vtc = VT + ((size_t)b * D_ + dcol) * N_;   // column dcol of V = row of VT

    v8f c = {0.f, 0.f, 0.f, 0.f, 0.f, 0.f, 0.f, 0.f};
#pragma unroll 4
    for (int m0 = 0; m0 < N_; m0 += 4) {
        v2f a  = *(const v2f*)(arow + m0 + 2 * lh);   // A: 16x4 fragment
        v2f bb = *(const v2f*)(vtc  + m0 + 2 * lh);   // B: 4x16 fragment via VT
        c = __builtin_amdgcn_wmma_f32_16x16x4_f32(false, a, false, bb,
                                                  (short)0, c, false, false);
    }
#pragma unroll
    for (int rr = 0; rr < 8; ++rr) {
        const int row = tn * 16 + rr + 8 * lh;
        const float v = c[rr];
        out[((size_t)b * N_ + row) * (2 * D_) + dcol] = v / (1.0f + fabsf(v));
    }
}

// ---------------- overwrite input-node rows with env input (pad == 0) ----------------
__global__ void k_env(const float* __restrict__ env, float* __restrict__ out) {
    const int idx = blockIdx.x * blockDim.x + threadIdx.x;   // B*64*128
    if (idx >= B_ * NUMIN * D_) return;
    const int b = idx / (NUMIN * D_);
    const int rem = idx % (NUMIN * D_);
    const int i = rem / D_, d = rem % D_;
    out[((size_t)b * N_ + i) * (2 * D_) + d] = env[(size_t)b * INDIM + i * D_ + d];
}

extern "C" void kernel_launch(void* const* d_in, const int* in_sizes, int n_in,
                              void* d_out, int out_size, void* d_ws, size_t ws_size,
                              hipStream_t stream) {
    const float* state  = (const float*)d_in[0];
    const float* output = (const float*)d_in[1];
    const float* w1     = (const float*)d_in[2];
    const float* w2     = (const float*)d_in[3];
    const float* w3     = (const float*)d_in[4];
    const float* A_ema  = (const float*)d_in[5];
    const float* env    = (const float*)d_in[6];
    const float* noise  = (const float*)d_in[7];
    float* out = (float*)d_out;

    float* ws   = (float*)d_ws;
    float* Q    = ws + Q_OFF;
    float* K    = ws + K_OFF;
    float* VT   = ws + VT_OFF;
    float* M    = ws + M_OFF;
    float* part = ws + P_OFF;
    float* scal = ws + SC_OFF;
    float* Aout = out + OUT_A_OFF;

    k_init<<<1, 32, 0, stream>>>(scal);

    // 1) per-node GEMVs + buntanh (bandwidth-dominant: ~805 MB streamed as b128)
    k_gemv<<<B_ * N_, 32, 0, stream>>>(state, w1, w2, w3, Q, K, out);

    // 2) raw_A = QK^T / sqrt(D) with f32 WMMA, fused sum-of-squares
    k_qk_wmma<<<B_ * 64 * 64, 32, 0, stream>>>(Q, K, M, scal);

    // 3) rms_norm + EMA + mask + max(.,0)+EPS
    k_ema<<<4096, 256, 0, stream>>>(A_ema, scal, M);

    // 4) sinkhorn #1
    for (int it = 0; it < ITERS; ++it) {
        k_rownorm<<<B_ * N_, 256, 0, stream>>>(M);
        k_colpart<<<B_ * RB_ * 4, 256, 0, stream>>>(M, part);
        k_colscale<<<B_ * RB_ * 4, 256, 0, stream>>>(M, part);
    }

    // 5) threshold + prep
    k_thresh<<<4096, 256, 0, stream>>>(M);

    // 6) sinkhorn #2
    for (int it = 0; it < ITERS; ++it) {
        k_rownorm<<<B_ * N_, 256, 0, stream>>>(M);
        k_colpart<<<B_ * RB_ * 4, 256, 0, stream>>>(M, part);
        k_colscale<<<B_ * RB_ * 4, 256, 0, stream>>>(M, part);
    }

    // 7) final A (masked) -> output, fused variance sums
    k_finalA<<<B_ * N_, 256, 0, stream>>>(M, Aout, scal);

    // 8) VT = transpose(output + vd*noise)
    k_vt<<<B_ * 32 * 4, dim3(32, 8), 0, stream>>>(output, noise, scal, VT);

    // 9) target = softsign(A@V) with f32 WMMA (K = 1024)
    k_av_wmma<<<B_ * 64 * 8, 32, 0, stream>>>(Aout, VT, out);

    // 10) env-input rows overwrite (pad == 0 so exact copy)
    k_env<<<(B_ * NUMIN * D_ + 255) / 256, 256, 0, stream>>>(env, out);
}